// GIN_NodeWeightEncoder_83760452207416
// MI455X (gfx1250) — compile-verified
//
#include <hip/hip_runtime.h>

#define N_NODES 100000
#define N_EDGES 1600000
#define DIM 64
#define OUTC 2
#define BN_EPS 1e-5f
#define LDS_STRIDE 68   // 16x64 f32 tile, row stride 68 floats -> banks rotate by 4

typedef __attribute__((ext_vector_type(2))) float v2f;
typedef __attribute__((ext_vector_type(8))) float v8f;

// ---------------------------------------------------------------- utilities

__global__ void zero_kernel(float* __restrict__ p, int n) {
    int i = blockIdx.x * blockDim.x + threadIdx.x;
    if (i < n) p[i] = 0.0f;
}

__global__ void copy_kernel(const float* __restrict__ src, float* __restrict__ dst, int n) {
    int i = blockIdx.x * blockDim.x + threadIdx.x;
    if (i < n) dst[i] = src[i];
}

// scatter-add: agg[dst] += feat[src], one thread per (edge, 4-float quad)
__global__ void scatter_kernel(const float* __restrict__ feat,
                               const long long* __restrict__ ei,
                               float* __restrict__ agg) {
    int gid = blockIdx.x * blockDim.x + threadIdx.x;
    int e = gid >> 4;
    int q = gid & 15;
    if (e >= N_EDGES) return;
    int s = (int)ei[e];
    int d = (int)ei[N_EDGES + e];
    const float4 v = *reinterpret_cast<const float4*>(feat + (size_t)s * DIM + q * 4);
    float* p = agg + (size_t)d * DIM + q * 4;
    atomicAdd(p + 0, v.x);
    atomicAdd(p + 1, v.y);
    atomicAdd(p + 2, v.z);
    atomicAdd(p + 3, v.w);
}

// ---------------------------------------------------------------- WMMA MLP

// One 64->64 Linear + bias + ReLU over the 16x64 LDS tile, result written back
// into the same tile. One wave (32 lanes), f32 WMMA, K=64 as 16 steps of K=4.
__device__ __forceinline__ void layer64(float* tile, const float* __restrict__ W,
                                        const float* __restrict__ bvec, int lane) {
    const int half = lane >> 4;    // A: 0 -> K pair {0,1}, 1 -> K pair {2,3}
    const int lrow = lane & 15;
    v8f acc[4] = {};
    for (int k = 0; k < 16; ++k) {
        const int kc = 4 * k + 2 * half;
        v2f a;
        a.x = tile[lrow * LDS_STRIDE + kc];
        a.y = tile[lrow * LDS_STRIDE + kc + 1];
        for (int t = 0; t < 4; ++t) {
            const int n = lrow + 16 * t;
            v2f b;
            b.x = W[kc * DIM + n];
            b.y = W[(kc + 1) * DIM + n];
            acc[t] = __builtin_amdgcn_wmma_f32_16x16x4_f32(
                false, a, false, b, (short)0, acc[t], false, false);
        }
    }
    __syncthreads();
    for (int t = 0; t < 4; ++t) {
        const int n = lrow + 16 * t;
        const float bias = bvec[n];
        for (int r = 0; r < 8; ++r) {
            float v = acc[t][r] + bias;
            v = v > 0.0f ? v : 0.0f;
            tile[(r + 8 * half) * LDS_STRIDE + n] = v;
        }
    }
    __syncthreads();
}

__device__ __forceinline__ void load_tile(float* tile, const float* __restrict__ A,
                                          int m0, int lane) {
    for (int idx = lane; idx < 256; idx += 32) {
        const int row = idx >> 4;
        const int c4 = (idx & 15) * 4;
        const float4 v = *reinterpret_cast<const float4*>(A + (size_t)(m0 + row) * DIM + c4);
        tile[row * LDS_STRIDE + c4 + 0] = v.x;
        tile[row * LDS_STRIDE + c4 + 1] = v.y;
        tile[row * LDS_STRIDE + c4 + 2] = v.z;
        tile[row * LDS_STRIDE + c4 + 3] = v.w;
    }
    __syncthreads();
}

// conv1 MLP: relu( relu(A@Wa+ba) @ Wb + bb ), store [N,64] + BN stats
__global__ __launch_bounds__(32) void gin_mlp64_kernel(
    const float* __restrict__ A, const float* __restrict__ Wa, const float* __restrict__ ba,
    const float* __restrict__ Wb, const float* __restrict__ bb,
    float* __restrict__ Hout, float* __restrict__ sum, float* __restrict__ sumsq) {
    __shared__ float tile[16 * LDS_STRIDE];
    const int lane = threadIdx.x;
    const int m0 = blockIdx.x * 16;
    load_tile(tile, A, m0, lane);
    layer64(tile, Wa, ba, lane);

    const int half = lane >> 4;
    const int lrow = lane & 15;
    v8f acc[4] = {};
    for (int k = 0; k < 16; ++k) {
        const int kc = 4 * k + 2 * half;
        v2f a;
        a.x = tile[lrow * LDS_STRIDE + kc];
        a.y = tile[lrow * LDS_STRIDE + kc + 1];
        for (int t = 0; t < 4; ++t) {
            const int n = lrow + 16 * t;
            v2f b;
            b.x = Wb[kc * DIM + n];
            b.y = Wb[(kc + 1) * DIM + n];
            acc[t] = __builtin_amdgcn_wmma_f32_16x16x4_f32(
                false, a, false, b, (short)0, acc[t], false, false);
        }
    }
    for (int t = 0; t < 4; ++t) {
        const int n = lrow + 16 * t;
        const float bias = bb[n];
        float s = 0.0f, s2 = 0.0f;
        for (int r = 0; r < 8; ++r) {
            float v = acc[t][r] + bias;
            v = v > 0.0f ? v : 0.0f;
            Hout[(size_t)(m0 + r + 8 * half) * DIM + n] = v;
            s += v;
            s2 += v * v;
        }
        atomicAdd(&sum[n], s);
        atomicAdd(&sumsq[n], s2);
    }
}

// conv5 MLP: relu( relu(A@Wa+ba) @ Wb[64x2] + bb ), store [N,2] + BN stats
__global__ __launch_bounds__(32) void gin_mlp2_kernel(
    const float* __restrict__ A, const float* __restrict__ Wa, const float* __restrict__ ba,
    const float* __restrict__ Wb, const float* __restrict__ bb,
    float* __restrict__ Out, float* __restrict__ sum, float* __restrict__ sumsq) {
    __shared__ float tile[16 * LDS_STRIDE];
    const int lane = threadIdx.x;
    const int m0 = blockIdx.x * 16;
    load_tile(tile, A, m0, lane);
    layer64(tile, Wa, ba, lane);

    const int half = lane >> 4;
    const int lrow = lane & 15;
    v8f acc = {};
    for (int k = 0; k < 16; ++k) {
        const int kc = 4 * k + 2 * half;
        v2f a;
        a.x = tile[lrow * LDS_STRIDE + kc];
        a.y = tile[lrow * LDS_STRIDE + kc + 1];
        v2f b;
        b.x = (lrow < OUTC) ? Wb[kc * OUTC + lrow] : 0.0f;
        b.y = (lrow < OUTC) ? Wb[(kc + 1) * OUTC + lrow] : 0.0f;
        acc = __builtin_amdgcn_wmma_f32_16x16x4_f32(
            false, a, false, b, (short)0, acc, false, false);
    }
    if (lrow < OUTC) {
        const float bias = bb[lrow];
        float s = 0.0f, s2 = 0.0f;
        for (int r = 0; r < 8; ++r) {
            float v = acc[r] + bias;
            v = v > 0.0f ? v : 0.0f;
            Out[(size_t)(m0 + r + 8 * half) * OUTC + lrow] = v;
            s += v;
            s2 += v * v;
        }
        atomicAdd(&sum[lrow], s);
        atomicAdd(&sumsq[lrow], s2);
    }
}

// ---------------------------------------------------------------- batchnorm

__global__ void finalize_stats_kernel(const float* __restrict__ sum,
                                      const float* __restrict__ sumsq,
                                      const float* __restrict__ gamma,
                                      const float* __restrict__ beta,
                                      float* __restrict__ scale, float* __restrict__ shift,
                                      int C, float invN) {
    int c = blockIdx.x * blockDim.x + threadIdx.x;
    if (c >= C) return;
    float mu = sum[c] * invN;
    float var = sumsq[c] * invN - mu * mu;
    float sc = gamma[c] * rsqrtf(var + BN_EPS);
    scale[c] = sc;
    shift[c] = beta[c] - mu * sc;
}

// h = h*scale+shift (in place); optionally also seed agg = h (GIN self term)
__global__ void apply_bn_kernel(float* __restrict__ h, float* __restrict__ agg,
                                const float* __restrict__ scale,
                                const float* __restrict__ shift, int total, int Cmask) {
    int i = blockIdx.x * blockDim.x + threadIdx.x;
    if (i >= total) return;
    int c = i & Cmask;
    float v = h[i] * scale[c] + shift[c];
    h[i] = v;
    if (agg) agg[i] = v;
}

// ---------------------------------------------------------------- launch

extern "C" void kernel_launch(void* const* d_in, const int* in_sizes, int n_in,
                              void* d_out, int out_size, void* d_ws, size_t ws_size,
                              hipStream_t stream) {
    const float* x      = (const float*)d_in[0];
    const long long* ei = (const long long*)d_in[1];
    const float* W1a    = (const float*)d_in[2];
    const float* b1a    = (const float*)d_in[3];
    const float* W1b    = (const float*)d_in[4];
    const float* b1b    = (const float*)d_in[5];
    const float* g1     = (const float*)d_in[6];
    const float* be1    = (const float*)d_in[7];
    const float* W5a    = (const float*)d_in[8];
    const float* b5a    = (const float*)d_in[9];
    const float* W5b    = (const float*)d_in[10];
    const float* b5b    = (const float*)d_in[11];
    const float* g5     = (const float*)d_in[12];
    const float* be5    = (const float*)d_in[13];
    float* out = (float*)d_out;

    const size_t nf = (size_t)N_NODES * DIM;           // 6.4M floats
    float* ws   = (float*)d_ws;
    float* bufA = ws;                                  // agg buffer (both convs)
    float* bufC = ws + nf;                             // h (pre-BN then post-BN)
    float* st   = ws + 2 * nf;                         // stats block (264 floats)
    float* sum1 = st;        float* sq1 = st + 64;
    float* sc1  = st + 128;  float* sh1 = st + 192;
    float* sum5 = st + 256;  float* sq5 = st + 258;
    float* sc5  = st + 260;  float* sh5 = st + 262;

    const int nfBlocks = (int)((nf + 255) / 256);                  // 25000
    const int edBlocks = (int)(((size_t)N_EDGES * 16 + 255) / 256);
    const int tiles    = N_NODES / 16;                             // 6250
    const float invN   = 1.0f / (float)N_NODES;

    // stats accumulators <- 0
    zero_kernel<<<2, 256, 0, stream>>>(st, 264);

    // ---- conv1: agg = x + scatter(x) ; MLP ; BN
    copy_kernel<<<nfBlocks, 256, 0, stream>>>(x, bufA, (int)nf);
    scatter_kernel<<<edBlocks, 256, 0, stream>>>(x, ei, bufA);
    gin_mlp64_kernel<<<tiles, 32, 0, stream>>>(bufA, W1a, b1a, W1b, b1b,
                                               bufC, sum1, sq1);
    finalize_stats_kernel<<<1, 64, 0, stream>>>(sum1, sq1, g1, be1, sc1, sh1, 64, invN);
    apply_bn_kernel<<<nfBlocks, 256, 0, stream>>>(bufC, bufA, sc1, sh1, (int)nf, DIM - 1);

    // ---- conv5: agg = h + scatter(h) ; MLP ; BN (in place on d_out)
    scatter_kernel<<<edBlocks, 256, 0, stream>>>(bufC, ei, bufA);
    gin_mlp2_kernel<<<tiles, 32, 0, stream>>>(bufA, W5a, b5a, W5b, b5b,
                                              out, sum5, sq5);
    finalize_stats_kernel<<<1, 2, 0, stream>>>(sum5, sq5, g5, be5, sc5, sh5, OUTC, invN);
    apply_bn_kernel<<<(out_size + 255) / 256, 256, 0, stream>>>(out, nullptr, sc5, sh5,
                                                                out_size, OUTC - 1);
}